// rpn_proposals_creator_26302379720848
// MI455X (gfx1250) — compile-verified
//
#include <hip/hip_runtime.h>
#include <stdint.h>

#define NBOX      147456
#define NUM_POST  300
#define IOU_THR   0.7f
#define NEGS      (-1e30f)
#define NWG       64
#define CHUNK     (NBOX / NWG)   // 2304 boxes per workgroup
#define BD        1024           // 32 wave32 waves per workgroup
#define NWAVE     (BD / 32)      // 32 -> second-stage reduce fits exactly in one wave

// ---- gfx1250 async global->LDS path (guarded; falls back to plain copies) ----
#if defined(__has_builtin)
#if __has_builtin(__builtin_amdgcn_global_load_async_to_lds_b128) && \
    __has_builtin(__builtin_amdgcn_s_wait_asynccnt)
#define HAVE_ASYNC_LDS 1
#endif
#endif

typedef __attribute__((address_space(1))) void as1_void;
typedef __attribute__((address_space(3))) void as3_void;
typedef int v4i __attribute__((vector_size(16)));               // matches builtin param
typedef __attribute__((address_space(1))) v4i as1_v4i;
typedef __attribute__((address_space(3))) v4i as3_v4i;

#ifdef HAVE_ASYNC_LDS
// async copy 16B: global (AS1) -> LDS (AS3).
__device__ __forceinline__ void async_cp16(const void* g, void* l) {
  __builtin_amdgcn_global_load_async_to_lds_b128(
      (as1_v4i*)(as1_void*)g, (as3_v4i*)(as3_void*)l, 0, 0);
}
#endif

// Monotonic float->uint mapping: order-preserving for all finite floats.
__device__ __forceinline__ unsigned fmono(float f) {
  unsigned b = __float_as_uint(f);
  return (b & 0x80000000u) ? ~b : (b | 0x80000000u);
}

// ---------------------------------------------------------------------------
// Kernel 1: decode + clip boxes -> AoS float4 box array + area array,
//           zero the per-iteration argmax slots and the global barrier state.
// ---------------------------------------------------------------------------
__global__ void decode_clip_kernel(const float4* __restrict__ anchors,
                                   const float4* __restrict__ deltas,
                                   const int* __restrict__ pH,
                                   const int* __restrict__ pW,
                                   float4* __restrict__ gbox,
                                   float*  __restrict__ gar,
                                   unsigned long long* __restrict__ slots,
                                   unsigned* __restrict__ bar) {
  const int i = blockIdx.x * blockDim.x + threadIdx.x;
  if (i < NUM_POST) slots[i] = 0ull;
  if (i == NUM_POST) { bar[0] = 0u; bar[1] = 0u; }
  if (i >= NBOX) return;

  const float Hf = (float)pH[0];
  const float Wf = (float)pW[0];
  const float4 a = anchors[i];
  const float4 d = deltas[i];

  const float ha  = a.z - a.x;
  const float wa  = a.w - a.y;
  const float cya = a.x + 0.5f * ha;
  const float cxa = a.y + 0.5f * wa;
  const float cy  = d.x * ha + cya;
  const float cx  = d.y * wa + cxa;
  const float h   = ha * expf(d.z);
  const float w   = wa * expf(d.w);

  const float y1 = fminf(fmaxf(cy - 0.5f * h, 0.f), Hf);
  const float x1 = fminf(fmaxf(cx - 0.5f * w, 0.f), Wf);
  const float y2 = fminf(fmaxf(cy + 0.5f * h, 0.f), Hf);
  const float x2 = fminf(fmaxf(cx + 0.5f * w, 0.f), Wf);

  gbox[i] = make_float4(y1, x1, y2, x2);
  gar[i]  = (y2 - y1) * (x2 - x1);
}

// ---------------------------------------------------------------------------
// Kernel 2: persistent cooperative NMS. 64 workgroups, each owns a 2304-box
// shard resident in LDS (AoS float4 boxes + areas + pre-monotonized scores).
// Per iteration: local LDS argmax (wave32 shuffle reduction) -> one
// global_atomic_max_u64 per WG -> device barrier -> broadcast winner ->
// LDS-resident suppression. One device-wide barrier per iteration.
// ---------------------------------------------------------------------------
__global__ void __launch_bounds__(BD, 1)
nms_persistent_kernel(const float* __restrict__ scores,
                      const float4* __restrict__ gbox,
                      const float* __restrict__ gar,
                      unsigned long long* __restrict__ slots,
                      unsigned* __restrict__ bar,
                      float* __restrict__ out) {
  extern __shared__ float4 smem[];
  float4*   sbox  = smem;                                 // CHUNK float4
  float*    sar   = (float*)(smem + CHUNK);               // CHUNK float
  unsigned* ssc_u = (unsigned*)(sar + CHUNK);             // CHUNK u32 (mono score)
  float*    ssc_f = (float*)ssc_u;                        // staging alias
  __shared__ unsigned long long red[NWAVE];

  const int base = blockIdx.x * CHUNK;
  const int lane = threadIdx.x & 31;
  const int wave = threadIdx.x >> 5;
  const unsigned SUPP   = fmono(NEGS);          // suppressed-entry key
  const unsigned THRESH = fmono(NEGS * 0.5f);   // validity threshold (mono space)

  // ---- Stage this WG's shard into LDS (async copy on gfx1250) ----
#ifdef HAVE_ASYNC_LDS
  for (int k = threadIdx.x; k < CHUNK; k += BD)
    async_cp16(gbox + base + k, sbox + k);                // one b128 per box
  for (int k = threadIdx.x * 4; k < CHUNK; k += BD * 4) {
    async_cp16(gar + base + k, sar + k);
    async_cp16(scores + base + k, ssc_f + k);
  }
  __builtin_amdgcn_s_wait_asynccnt(0);
#else
  for (int k = threadIdx.x; k < CHUNK; k += BD) {
    sbox[k]  = gbox[base + k];
    sar[k]   = gar[base + k];
    ssc_f[k] = scores[base + k];
  }
#endif
  __syncthreads();
  // pre-monotonize scores in place (each element touched by exactly one thread)
  for (int k = threadIdx.x; k < CHUNK; k += BD) ssc_u[k] = fmono(ssc_f[k]);
  __syncthreads();

  for (int iter = 0; iter < NUM_POST; ++iter) {
    // ---- local argmax over LDS shard; key = (mono score, first-index) ----
    unsigned long long best = 0ull;
    for (int k = threadIdx.x; k < CHUNK; k += BD) {
      unsigned long long key =
          ((unsigned long long)ssc_u[k] << 32) |
          (unsigned long long)(0xFFFFFFFFu - (unsigned)(base + k));
      best = (key > best) ? key : best;
    }
#pragma unroll
    for (int off = 16; off > 0; off >>= 1) {
      unsigned long long o = __shfl_xor(best, off, 32);
      best = (o > best) ? o : best;
    }
    if (lane == 0) red[wave] = best;
    __syncthreads();
    if (wave == 0) {
      best = red[lane];  // NWAVE == 32: one partial per lane
#pragma unroll
      for (int off = 16; off > 0; off >>= 1) {
        unsigned long long o = __shfl_xor(best, off, 32);
        best = (o > best) ? o : best;
      }
      if (lane == 0) atomicMax(&slots[iter], best);  // global_atomic_max_u64
    }

    // ---- device-wide barrier: gen counter reaches iter+1 ----
    __threadfence();
    if (threadIdx.x == 0) {
      unsigned arrived = __hip_atomic_fetch_add(&bar[0], 1u, __ATOMIC_ACQ_REL,
                                                __HIP_MEMORY_SCOPE_AGENT);
      if (arrived == NWG - 1) {
        __hip_atomic_store(&bar[0], 0u, __ATOMIC_RELAXED, __HIP_MEMORY_SCOPE_AGENT);
        __hip_atomic_fetch_add(&bar[1], 1u, __ATOMIC_ACQ_REL, __HIP_MEMORY_SCOPE_AGENT);
      } else {
        while (__hip_atomic_load(&bar[1], __ATOMIC_ACQUIRE,
                                 __HIP_MEMORY_SCOPE_AGENT) < (unsigned)(iter + 1)) {
          __builtin_amdgcn_s_sleep(2);
        }
      }
    }
    __syncthreads();
    __threadfence();

    // ---- read winner (final: all WGs deposited before barrier released) ----
    const unsigned long long key =
        __hip_atomic_load(&slots[iter], __ATOMIC_RELAXED, __HIP_MEMORY_SCOPE_AGENT);
    const int  j     = (int)(0xFFFFFFFFu - (unsigned)(key & 0xFFFFFFFFull));
    const bool valid = ((unsigned)(key >> 32)) > THRESH;

    const float4 bj = gbox[j];     // uniform address -> scalar broadcast loads
    const float  ba = gar[j];

    if (blockIdx.x == 0 && threadIdx.x == 0) {
      ((float4*)out)[iter] = valid ? bj : make_float4(0.f, 0.f, 0.f, 0.f);
    }

    // ---- suppress within LDS-resident shard (1x b128 + 1x b32 per box) ----
    if (valid) {
      for (int k = threadIdx.x; k < CHUNK; k += BD) {
        const float4 b = sbox[k];
        const float yy1 = fmaxf(bj.x, b.x);
        const float xx1 = fmaxf(bj.y, b.y);
        const float yy2 = fminf(bj.z, b.z);
        const float xx2 = fminf(bj.w, b.w);
        const float inter = fmaxf(yy2 - yy1, 0.f) * fmaxf(xx2 - xx1, 0.f);
        const float iou = inter / (ba + sar[k] - inter + 1e-8f);
        if (iou > IOU_THR) ssc_u[k] = SUPP;
      }
    }
    __syncthreads();  // shard scores final before next iteration's argmax
  }
}

// ---------------------------------------------------------------------------
extern "C" void kernel_launch(void* const* d_in, const int* in_sizes, int n_in,
                              void* d_out, int out_size, void* d_ws, size_t ws_size,
                              hipStream_t stream) {
  (void)in_sizes; (void)n_in; (void)out_size; (void)ws_size;

  const float* scores  = (const float*)d_in[0];   // rpn_scores      [N]
  const float* deltas  = (const float*)d_in[1];   // rpn_coordinates [N,4]
  const float* anchors = (const float*)d_in[2];   // anchors         [N,4]
  const int*   pH      = (const int*)d_in[3];     // image_h (scalar)
  const int*   pW      = (const int*)d_in[4];     // image_w (scalar)

  float4* gbox = (float4*)d_ws;                                   // N float4
  float*  gar  = (float*)(gbox + NBOX);                           // N float
  unsigned long long* slots = (unsigned long long*)(gar + NBOX);  // NUM_POST u64
  unsigned* bar = (unsigned*)(slots + NUM_POST);                  // 2 u32

  decode_clip_kernel<<<NBOX / 256, 256, 0, stream>>>(
      (const float4*)anchors, (const float4*)deltas, pH, pW,
      gbox, gar, slots, bar);

  // LDS: CHUNK*16 (boxes) + CHUNK*4 (areas) + CHUNK*4 (mono scores) = 55,296 B
  const size_t shmem = (size_t)CHUNK * (sizeof(float4) + 2 * sizeof(float));
  nms_persistent_kernel<<<NWG, BD, shmem, stream>>>(
      scores, gbox, gar, slots, bar, (float*)d_out);
}